// GraphPoolingModel_layer_3_51616916963376
// MI455X (gfx1250) — compile-verified
//
#include <hip/hip_runtime.h>

typedef __attribute__((ext_vector_type(2))) float v2f;
typedef __attribute__((ext_vector_type(8))) float v8f;

#define HD 128   // hidden/feature dim
#define GNUM 64  // graphs

// ---- monotone float<->uint encoding for atomic max over signed floats ----
__device__ __forceinline__ unsigned enc_f(float f) {
  unsigned u = __float_as_uint(f);
  return (u & 0x80000000u) ? ~u : (u | 0x80000000u);
}
__device__ __forceinline__ float dec_f(unsigned u) {
  unsigned v = (u & 0x80000000u) ? (u & 0x7fffffffu) : ~u;
  return __uint_as_float(v);
}

__global__ void fill_u32_kernel(unsigned* __restrict__ p, unsigned v, long long n) {
  long long i = (long long)blockIdx.x * blockDim.x + threadIdx.x;
  if (i < n) p[i] = v;
}

__global__ void deg_kernel(const int* __restrict__ ei, float* __restrict__ deg, int E) {
  int i = blockIdx.x * blockDim.x + threadIdx.x;
  if (i < E) atomicAdd(&deg[ei[E + i]], 1.0f);
}

__global__ void dinv_kernel(const float* __restrict__ deg, float* __restrict__ dinv, int n) {
  int i = blockIdx.x * blockDim.x + threadIdx.x;
  if (i < n) dinv[i] = rsqrtf(deg[i] + 1.0f);
}

// C[M,NC] = A[M,K] @ B[K,NC] (+ bias per column, optional relu)
// fp32 WMMA 16x16x4: one wave per 16x16 output tile, K stepped by 4.
// Requires M%16==0, NC%16==0, K%4==0 (all shapes here satisfy this).
__global__ void gemm_wmma_f32_kernel(const float* __restrict__ A,
                                     const float* __restrict__ B,
                                     const float* __restrict__ bias,
                                     float* __restrict__ C,
                                     int M, int K, int NC, int relu) {
  int wave = (int)(((long long)blockIdx.x * blockDim.x + threadIdx.x) >> 5);
  int lane = threadIdx.x & 31;
  int tilesN = NC >> 4;
  int total = (M >> 4) * tilesN;
  if (wave >= total) return;               // wave-uniform: EXEC stays all-1s for WMMA
  int tm = wave / tilesN, tn = wave - tm * tilesN;
  int row0 = tm << 4, col0 = tn << 4;
  int l = lane & 15, hi = lane >> 4;

  const float* Arow = A + (size_t)(row0 + l) * K;   // lane index = M for A frags
  const float* Bcol = B + (size_t)col0 + l;         // lane index = N for B frags
  v8f acc = {};
  for (int k = 0; k < K; k += 4) {
    v2f a, b;
    a.x = Arow[k + 2 * hi];                         // A[m][k+2hi]
    a.y = Arow[k + 2 * hi + 1];                     // A[m][k+2hi+1]
    b.x = Bcol[(size_t)(k + 2 * hi) * NC];          // B[k+2hi][n]
    b.y = Bcol[(size_t)(k + 2 * hi + 1) * NC];      // B[k+2hi+1][n]
    acc = __builtin_amdgcn_wmma_f32_16x16x4_f32(false, a, false, b, (short)0, acc,
                                                false, false);
  }
  float bl = bias ? bias[col0 + l] : 0.0f;
#pragma unroll
  for (int j = 0; j < 8; ++j) {
    float v = acc[j] + bl;
    if (relu) v = fmaxf(v, 0.0f);
    C[(size_t)(row0 + j + 8 * hi) * NC + col0 + l] = v;   // D row = j + 8*hi
  }
}

// One wave per edge: agg[dst] += h[src] * dinv[src]*dinv[dst]   (128 feats, 4/lane)
__global__ void scatter_kernel(const float* __restrict__ h, const int* __restrict__ ei,
                               const float* __restrict__ dinv, float* __restrict__ agg,
                               int E) {
  int wave = (int)(((long long)blockIdx.x * blockDim.x + threadIdx.x) >> 5);
  int lane = threadIdx.x & 31;
  if (wave >= E) return;
  int s = ei[wave];
  int d = ei[E + wave];
  float c = dinv[s] * dinv[d];
  const float4 hv = *(const float4*)(h + (size_t)s * HD + lane * 4);
  float* ap = agg + (size_t)d * HD + lane * 4;
  atomicAdd(ap + 0, hv.x * c);
  atomicAdd(ap + 1, hv.y * c);
  atomicAdd(ap + 2, hv.z * c);
  atomicAdd(ap + 3, hv.w * c);
}

// One wave per node: v = agg + h*dinv^2 + b ; relu ; layernorm(g,be) -> out
__global__ void combine_ln_kernel(const float* __restrict__ h, const float* __restrict__ agg,
                                  const float* __restrict__ dinv, const float* __restrict__ bias,
                                  const float* __restrict__ gamma, const float* __restrict__ beta,
                                  float* __restrict__ out, int n) {
  int wave = (int)(((long long)blockIdx.x * blockDim.x + threadIdx.x) >> 5);
  int lane = threadIdx.x & 31;
  if (wave >= n) return;
  float di = dinv[wave];
  float selfc = di * di;
  size_t base = (size_t)wave * HD + lane * 4;
  const float4 hv = *(const float4*)(h + base);
  const float4 av = *(const float4*)(agg + base);
  int f = lane * 4;
  float v0 = fmaxf(av.x + hv.x * selfc + bias[f + 0], 0.0f);
  float v1 = fmaxf(av.y + hv.y * selfc + bias[f + 1], 0.0f);
  float v2 = fmaxf(av.z + hv.z * selfc + bias[f + 2], 0.0f);
  float v3 = fmaxf(av.w + hv.w * selfc + bias[f + 3], 0.0f);
  float s  = v0 + v1 + v2 + v3;
  float s2 = v0 * v0 + v1 * v1 + v2 * v2 + v3 * v3;
#pragma unroll
  for (int off = 16; off > 0; off >>= 1) {     // wave32 reduction
    s  += __shfl_xor(s,  off, 32);
    s2 += __shfl_xor(s2, off, 32);
  }
  const float inv = 1.0f / (float)HD;
  float mean = s * inv;
  float var  = s2 * inv - mean * mean;
  float r = rsqrtf(var + 1e-5f);
  out[base + 0] = (v0 - mean) * r * gamma[f + 0] + beta[f + 0];
  out[base + 1] = (v1 - mean) * r * gamma[f + 1] + beta[f + 1];
  out[base + 2] = (v2 - mean) * r * gamma[f + 2] + beta[f + 2];
  out[base + 3] = (v3 - mean) * r * gamma[f + 3] + beta[f + 3];
}

// One wave per node: atomic add into sum[g], atomic max into maxu[g], count nodes
__global__ void pool_kernel(const float* __restrict__ h, const int* __restrict__ batch,
                            float* __restrict__ sum, int* __restrict__ cnt,
                            unsigned* __restrict__ maxu, int n) {
  int wave = (int)(((long long)blockIdx.x * blockDim.x + threadIdx.x) >> 5);
  int lane = threadIdx.x & 31;
  if (wave >= n) return;
  int g = batch[wave];
  const float4 hv = *(const float4*)(h + (size_t)wave * HD + lane * 4);
  size_t gb = (size_t)g * HD + lane * 4;
  atomicAdd(sum + gb + 0, hv.x);
  atomicAdd(sum + gb + 1, hv.y);
  atomicAdd(sum + gb + 2, hv.z);
  atomicAdd(sum + gb + 3, hv.w);
  atomicMax(maxu + gb + 0, enc_f(hv.x));
  atomicMax(maxu + gb + 1, enc_f(hv.y));
  atomicMax(maxu + gb + 2, enc_f(hv.z));
  atomicMax(maxu + gb + 3, enc_f(hv.w));
  if (lane == 0) atomicAdd(&cnt[g], 1);
}

// z[g] = concat(mean, add, max)  -> [G, 3H]
__global__ void build_z_kernel(const float* __restrict__ sum, const int* __restrict__ cnt,
                               const unsigned* __restrict__ maxu, float* __restrict__ z) {
  int i = blockIdx.x * blockDim.x + threadIdx.x;
  if (i >= GNUM * HD) return;
  int g = i >> 7, f = i & (HD - 1);
  float s = sum[i];
  int c = cnt[g];
  float mean = s / fmaxf((float)c, 1.0f);
  float mx = (c > 0) ? dec_f(maxu[i]) : 0.0f;
  float* zr = z + (size_t)g * (3 * HD);
  zr[f] = mean;
  zr[HD + f] = s;
  zr[2 * HD + f] = mx;
}

extern "C" void kernel_launch(void* const* d_in, const int* in_sizes, int n_in,
                              void* d_out, int out_size, void* d_ws, size_t ws_size,
                              hipStream_t stream) {
  const float* x  = (const float*)d_in[0];
  const int* ei   = (const int*)d_in[1];
  const int* batch= (const int*)d_in[2];
  const float* W1 = (const float*)d_in[3];  const float* b1  = (const float*)d_in[4];
  const float* W2 = (const float*)d_in[5];  const float* b2  = (const float*)d_in[6];
  const float* W3 = (const float*)d_in[7];  const float* b3  = (const float*)d_in[8];
  const float* g1 = (const float*)d_in[9];  const float* be1 = (const float*)d_in[10];
  const float* g2 = (const float*)d_in[11]; const float* be2 = (const float*)d_in[12];
  const float* g3 = (const float*)d_in[13]; const float* be3 = (const float*)d_in[14];
  const float* fW1= (const float*)d_in[15]; const float* fb1 = (const float*)d_in[16];
  const float* fW2= (const float*)d_in[17]; const float* fb2 = (const float*)d_in[18];
  const float* fW3= (const float*)d_in[19]; const float* fb3 = (const float*)d_in[20];
  float* outp = (float*)d_out;

  const int N = in_sizes[0] / HD;       // 50000
  const int E = in_sizes[1] / 2;        // 1600000

  // ---- workspace carve-up (all chunks 16B aligned) ----
  char* w = (char*)d_ws;
  float*    buf_h   = (float*)w;                 w += (size_t)N * HD * 4;
  float*    buf_agg = (float*)w;                 w += (size_t)N * HD * 4;
  float*    buf_out = (float*)w;                 w += (size_t)N * HD * 4;
  float*    deg     = (float*)w;                 w += (size_t)N * 4;
  float*    dinv    = (float*)w;                 w += (size_t)N * 4;
  float*    psum    = (float*)w;                 w += (size_t)GNUM * HD * 4;
  unsigned* pmax    = (unsigned*)w;              w += (size_t)GNUM * HD * 4;
  int*      pcnt    = (int*)w;                   w += (size_t)GNUM * 4;
  float*    z       = (float*)w;                 w += (size_t)GNUM * 3 * HD * 4;
  float*    t1      = (float*)w;                 w += (size_t)GNUM * HD * 4;
  float*    t2      = (float*)w;                 w += (size_t)GNUM * (HD / 2) * 4;

  const int TB = 256;
  auto blk1 = [](long long n) { return (int)((n + 255) / 256); };     // thread-per-elem
  auto blkw = [](long long w_) { return (int)((w_ * 32 + 255) / 256); }; // wave-per-item

  // degrees (with self-loop) and rsqrt
  hipLaunchKernelGGL(fill_u32_kernel, dim3(blk1(N)), dim3(TB), 0, stream,
                     (unsigned*)deg, 0u, (long long)N);
  hipLaunchKernelGGL(deg_kernel, dim3(blk1(E)), dim3(TB), 0, stream, ei, deg, E);
  hipLaunchKernelGGL(dinv_kernel, dim3(blk1(N)), dim3(TB), 0, stream, deg, dinv, N);

  const float* layer_in = x;
  const float* Ws[3]  = {W1, W2, W3};
  const float* bs[3]  = {b1, b2, b3};
  const float* gs[3]  = {g1, g2, g3};
  const float* bes[3] = {be1, be2, be3};

  for (int L = 0; L < 3; ++L) {
    long long gemm_waves = (long long)(N / 16) * (HD / 16);
    hipLaunchKernelGGL(gemm_wmma_f32_kernel, dim3(blkw(gemm_waves)), dim3(TB), 0, stream,
                       layer_in, Ws[L], (const float*)nullptr, buf_h, N, HD, HD, 0);
    hipLaunchKernelGGL(fill_u32_kernel, dim3(blk1((long long)N * HD)), dim3(TB), 0, stream,
                       (unsigned*)buf_agg, 0u, (long long)N * HD);
    hipLaunchKernelGGL(scatter_kernel, dim3(blkw(E)), dim3(TB), 0, stream,
                       buf_h, ei, dinv, buf_agg, E);
    hipLaunchKernelGGL(combine_ln_kernel, dim3(blkw(N)), dim3(TB), 0, stream,
                       buf_h, buf_agg, dinv, bs[L], gs[L], bes[L], buf_out, N);
    layer_in = buf_out;
  }

  // pooling
  hipLaunchKernelGGL(fill_u32_kernel, dim3(blk1(GNUM * HD)), dim3(TB), 0, stream,
                     (unsigned*)psum, 0u, (long long)GNUM * HD);
  hipLaunchKernelGGL(fill_u32_kernel, dim3(blk1(GNUM * HD)), dim3(TB), 0, stream,
                     pmax, 0u, (long long)GNUM * HD);
  hipLaunchKernelGGL(fill_u32_kernel, dim3(1), dim3(TB), 0, stream,
                     (unsigned*)pcnt, 0u, (long long)GNUM);
  hipLaunchKernelGGL(pool_kernel, dim3(blkw(N)), dim3(TB), 0, stream,
                     buf_out, batch, psum, pcnt, pmax, N);
  hipLaunchKernelGGL(build_z_kernel, dim3(blk1(GNUM * HD)), dim3(TB), 0, stream,
                     psum, pcnt, pmax, z);

  // MLP head: [64,384]@[384,128] -> relu -> [64,128]@[128,64] -> relu -> [64,64]@[64,16]
  hipLaunchKernelGGL(gemm_wmma_f32_kernel, dim3(blkw((GNUM / 16) * (HD / 16))), dim3(TB), 0,
                     stream, z, fW1, fb1, t1, GNUM, 3 * HD, HD, 1);
  hipLaunchKernelGGL(gemm_wmma_f32_kernel, dim3(blkw((GNUM / 16) * (HD / 32))), dim3(TB), 0,
                     stream, t1, fW2, fb2, t2, GNUM, HD, HD / 2, 1);
  hipLaunchKernelGGL(gemm_wmma_f32_kernel, dim3(blkw(GNUM / 16)), dim3(TB), 0,
                     stream, t2, fW3, fb3, outp, GNUM, HD / 2, 16, 0);
}